// XSMMTriangleMultiplication_49460843380749
// MI455X (gfx1250) — compile-verified
//
#include <hip/hip_runtime.h>
#include <hip/hip_bf16.h>

// ---------------------------------------------------------------------------
// Triangle multiplication (outgoing) for MI455X / gfx1250, fp32 WMMA path.
// N=512, C=128.
// ws layout: a[c][p] | b[c][p] | g[c][p] | tri[c][p] | wcat[640][128]
// ---------------------------------------------------------------------------

#define Nn 512
#define Cc 128
#define NP (Nn * Nn)

typedef __attribute__((ext_vector_type(2))) float v2f;
typedef __attribute__((ext_vector_type(8))) float v8f;
typedef __attribute__((address_space(3))) float lds_f;

// V_WMMA_F32_16X16X4_F32: D(16x16 f32) = A(16x4 f32) x B(4x16 f32) + C
__device__ __forceinline__ v8f wmma4(v2f a, v2f b, v8f c) {
    return __builtin_amdgcn_wmma_f32_16x16x4_f32(
        false, a, false, b, (short)0, c, false, false);
}

__device__ __forceinline__ float sigmoidf(float x) {
    return 1.0f / (1.0f + __expf(-x));
}

// Async copy 16B/lane global -> LDS, tracked by ASYNCcnt (no VGPR round-trip).
__device__ __forceinline__ void async_ld128(const float* g, float* l) {
    asm volatile("global_load_async_to_lds_b128 %0, %1, off"
                 :: "v"((lds_f*)l), "v"(g)
                 : "memory");
}
__device__ __forceinline__ void wait_async0() {
    asm volatile("s_wait_asynccnt 0x0" ::: "memory");
}

// ---------------------------------------------------------------------------
// Kernel 0: pack proj_w(256x128) | gate_w(256x128) | gating_w(128x128) into
// one contiguous 640x128 row-major matrix so k1's B-fragment loads come from
// a single kernel-arg base pointer (=> global_load_b64 GVS, never flat).
// ---------------------------------------------------------------------------
__global__ __launch_bounds__(256) void k0_pack(
    const float* __restrict__ projw, const float* __restrict__ gatew,
    const float* __restrict__ gatingw, float* __restrict__ wcat)
{
    int q = blockIdx.x * 256 + threadIdx.x;      // float4 index, 0..20479
    float4 v;
    if (q < 256 * Cc / 4)            v = *(const float4*)(projw  + q * 4);
    else if (q < 512 * Cc / 4)       v = *(const float4*)(gatew  + (q - 256 * Cc / 4) * 4);
    else                             v = *(const float4*)(gatingw + (q - 512 * Cc / 4) * 4);
    *(float4*)(wcat + q * 4) = v;
}

// ---------------------------------------------------------------------------
// Kernel 1: LN1 + [proj | gate | gating] GEMM + sigmoid gating + even/odd
// split.  One WG = 16 positions p = i*N+j, 4 waves x 10 n-tiles of wcat.
// Writes a,b,g channel-major: ws[c*NP + p].
// ---------------------------------------------------------------------------
__global__ __launch_bounds__(128) void k1_ln_proj(
    const float* __restrict__ pair,  const float* __restrict__ mask,
    const float* __restrict__ ln1w,  const float* __restrict__ ln1b,
    const float* __restrict__ wcat,
    float* __restrict__ a_ws, float* __restrict__ b_ws, float* __restrict__ g_ws)
{
    __shared__ float y[16][Cc + 4];   // stride 132 floats (528B, 16B aligned)
    __shared__ float raw[16][648];    // proj[0:256) gate[256:512) gating[512:640)

    const int t    = threadIdx.x;
    const int lane = t & 31;
    const int wave = t >> 5;
    const int p0   = blockIdx.x * 16;

    // ---- load 16 x 128 input tile (coalesced float4) ----
    #pragma unroll
    for (int it = 0; it < 4; ++it) {
        int idx = t + 128 * it;              // 0..511 float4 slots
        int row = idx >> 5;                  // 16 rows
        int c4  = (idx & 31) * 4;
        *(float4*)&y[row][c4] =
            *(const float4*)(pair + (size_t)(p0 + row) * Cc + c4);
    }
    __syncthreads();

    // ---- LayerNorm (wave w handles rows 4w..4w+3, shuffle reduction) ----
    #pragma unroll
    for (int rr = 0; rr < 4; ++rr) {
        int row = wave * 4 + rr;
        float vals[4];
        float s = 0.0f;
        #pragma unroll
        for (int q = 0; q < 4; ++q) { vals[q] = y[row][lane + 32 * q]; s += vals[q]; }
        #pragma unroll
        for (int off = 16; off; off >>= 1) s += __shfl_xor(s, off, 32);
        float mu = s * (1.0f / Cc);
        float v = 0.0f;
        #pragma unroll
        for (int q = 0; q < 4; ++q) { float d = vals[q] - mu; v += d * d; }
        #pragma unroll
        for (int off = 16; off; off >>= 1) v += __shfl_xor(v, off, 32);
        float rstd = __frsqrt_rn(v * (1.0f / Cc) + 1e-5f);
        #pragma unroll
        for (int q = 0; q < 4; ++q) {
            int c = lane + 32 * q;
            y[row][c] = (vals[q] - mu) * rstd * ln1w[c] + ln1b[c];
        }
    }
    __syncthreads();

    // ---- WMMA GEMM: 16 pos x 640 outputs, K=128; 10 n-tiles per wave ----
    const int mrow = lane & 15;          // A/B row index within tile
    const int ks   = (lane >> 4) * 2;    // K pair select (lanes 16-31 -> K=2,3)

    // Per-lane row base into wcat; nt/k offsets fold into the 24-bit ioffset.
    const float* wrow = wcat + (size_t)(wave * 160 + mrow) * Cc + ks;

    v8f acc[10];
    #pragma unroll
    for (int nt = 0; nt < 10; ++nt) acc[nt] = (v8f){};

    for (int k0 = 0; k0 < Cc; k0 += 4) {
        v2f af = *(const v2f*)&y[mrow][k0 + ks];
        v2f bf[10];
        #pragma unroll
        for (int nt = 0; nt < 10; ++nt)
            bf[nt] = *(const v2f*)(wrow + nt * 16 * Cc + k0);
        #pragma unroll
        for (int nt = 0; nt < 10; ++nt)
            acc[nt] = wmma4(af, bf[nt], acc[nt]);
    }

    // ---- spill raw results to LDS (C/D layout: m = (lane>>4)*8 + r) ----
    const int mbase = (lane >> 4) * 8;
    #pragma unroll
    for (int nt = 0; nt < 10; ++nt) {
        int col = wave * 160 + nt * 16 + mrow;
        #pragma unroll
        for (int r = 0; r < 8; ++r) raw[mbase + r][col] = acc[nt][r];
    }
    __syncthreads();

    // ---- epilogue: mask * sigmoid gating, even/odd split, channel-major ----
    const int m  = t & 15;
    const size_t p = (size_t)p0 + m;
    const float mk = mask[p];
    #pragma unroll
    for (int pass = 0; pass < 16; ++pass) {
        int c = pass * 8 + (t >> 4);         // 0..127
        float av = raw[m][2 * c]     * mk * sigmoidf(raw[m][256 + 2 * c]);
        float bv = raw[m][2 * c + 1] * mk * sigmoidf(raw[m][256 + 2 * c + 1]);
        float gv = sigmoidf(raw[m][512 + c]);
        a_ws[(size_t)c * NP + p] = av;
        b_ws[(size_t)c * NP + p] = bv;
        g_ws[(size_t)c * NP + p] = gv;
    }
}

// ---------------------------------------------------------------------------
// Kernel 2: tri[c] = A[c] (512x512) @ B[c]^T (NT GEMM), batched over 128 c.
// Grid (8,8,128): one 64x64 output tile per WG; 4 waves x (2x2) WMMA tiles.
// Double-buffered: next k-panel streamed global->LDS with
// GLOBAL_LOAD_ASYNC_TO_LDS_B128 (ASYNCcnt) while WMMAs consume the current
// panel; s_wait_asynccnt 0 + barrier at chunk boundaries.
// ---------------------------------------------------------------------------
__global__ __launch_bounds__(128) void k2_tri(
    const float* __restrict__ a_ws, const float* __restrict__ b_ws,
    float* __restrict__ tri_ws)
{
    __shared__ float As[2][64][68];
    __shared__ float Bs[2][64][68];

    const int t    = threadIdx.x;
    const int lane = t & 31;
    const int wave = t >> 5;
    const int c    = blockIdx.z;
    const int i0   = blockIdx.x * 64;
    const int j0   = blockIdx.y * 64;

    const float* A = a_ws + (size_t)c * NP;
    const float* B = b_ws + (size_t)c * NP;

    const int mw   = (wave & 1) * 32;    // wave's 32x32 sub-tile
    const int nw   = (wave >> 1) * 32;
    const int mrow = lane & 15;
    const int ks   = (lane >> 4) * 2;

    // this thread's 8 (row, col4) slots for panel staging
    const int srow = t >> 4;             // 0..7 base row group
    const int sc4  = (t & 15) * 4;       // column within panel

    v8f acc[2][2];
    #pragma unroll
    for (int a0 = 0; a0 < 2; ++a0)
        #pragma unroll
        for (int b0 = 0; b0 < 2; ++b0) acc[a0][b0] = (v8f){};

    // ---- prologue: async-stage k-chunk 0 into buffer 0 ----
    #pragma unroll
    for (int it = 0; it < 8; ++it) {
        int row = srow + 8 * it;         // 0..63
        async_ld128(A + (size_t)(i0 + row) * Nn + sc4, &As[0][row][sc4]);
        async_ld128(B + (size_t)(j0 + row) * Nn + sc4, &Bs[0][row][sc4]);
    }

    for (int kc = 0; kc < Nn / 64; ++kc) {
        const int cur = kc & 1;
        wait_async0();                   // my panel-kc loads landed in LDS
        __syncthreads();                 // everyone's landed; prior reads done

        if (kc + 1 < Nn / 64) {          // prefetch panel kc+1 into other buf
            int kg = (kc + 1) * 64;
            #pragma unroll
            for (int it = 0; it < 8; ++it) {
                int row = srow + 8 * it;
                async_ld128(A + (size_t)(i0 + row) * Nn + kg + sc4,
                            &As[cur ^ 1][row][sc4]);
                async_ld128(B + (size_t)(j0 + row) * Nn + kg + sc4,
                            &Bs[cur ^ 1][row][sc4]);
            }
        }

        #pragma unroll 4
        for (int k = 0; k < 64; k += 4) {
            v2f a0 = *(const v2f*)&As[cur][mw + mrow][k + ks];
            v2f a1 = *(const v2f*)&As[cur][mw + 16 + mrow][k + ks];
            v2f b0 = *(const v2f*)&Bs[cur][nw + mrow][k + ks];
            v2f b1 = *(const v2f*)&Bs[cur][nw + 16 + mrow][k + ks];
            acc[0][0] = wmma4(a0, b0, acc[0][0]);
            acc[0][1] = wmma4(a0, b1, acc[0][1]);
            acc[1][0] = wmma4(a1, b0, acc[1][0]);
            acc[1][1] = wmma4(a1, b1, acc[1][1]);
        }
        __syncthreads();                 // reads of cur done before reuse
    }

    // store channel-major tri[c][i][j] (contiguous across lanes 0-15)
    float* T  = tri_ws + (size_t)c * NP;
    const int mbase = (lane >> 4) * 8;
    const int nl    = lane & 15;
    #pragma unroll
    for (int mt = 0; mt < 2; ++mt)
        #pragma unroll
        for (int nt = 0; nt < 2; ++nt)
            #pragma unroll
            for (int r = 0; r < 8; ++r)
                T[(size_t)(i0 + mw + mt * 16 + mbase + r) * Nn
                  + (j0 + nw + nt * 16 + nl)] = acc[mt][nt][r];
}

// ---------------------------------------------------------------------------
// Kernel 3: LN2(tri) @ out_w^T * g, output (i,j,c).
// ---------------------------------------------------------------------------
__global__ __launch_bounds__(128) void k3_out(
    const float* __restrict__ tri_ws, const float* __restrict__ g_ws,
    const float* __restrict__ ln2w,   const float* __restrict__ ln2b,
    const float* __restrict__ outw,   float* __restrict__ out)
{
    __shared__ float Ts[Cc][68];
    __shared__ float Gs[Cc][68];

    const int t    = threadIdx.x;
    const int lane = t & 31;
    const int wave = t >> 5;
    const int i    = blockIdx.x;
    const int j0   = blockIdx.y * 64;

    // stage 128(c) x 64(j) tiles of tri and g (coalesced float4 rows)
    #pragma unroll
    for (int it = 0; it < 16; ++it) {
        int idx = t + 128 * it;              // 0..2047
        int row = idx >> 4;                  // channel 0..127
        int c4  = (idx & 15) * 4;            // j offset
        size_t ga = (size_t)row * NP + (size_t)i * Nn + j0 + c4;
        *(float4*)&Ts[row][c4] = *(const float4*)(tri_ws + ga);
        *(float4*)&Gs[row][c4] = *(const float4*)(g_ws + ga);
    }
    __syncthreads();

    // LN2 over channels for each of the 64 j columns (in place)
    if (t < 64) {
        float s = 0.0f;
        for (int ch = 0; ch < Cc; ++ch) s += Ts[ch][t];
        float mu = s * (1.0f / Cc);
        float v = 0.0f;
        for (int ch = 0; ch < Cc; ++ch) { float d = Ts[ch][t] - mu; v += d * d; }
        float rstd = __frsqrt_rn(v * (1.0f / Cc) + 1e-5f);
        for (int ch = 0; ch < Cc; ++ch)
            Ts[ch][t] = (Ts[ch][t] - mu) * rstd * ln2w[ch] + ln2b[ch];
    }
    __syncthreads();

    // WMMA GEMM: 16 j-rows per wave x 128 c_out, K = 128 channels
    const int jb = wave * 16;
    const int nl = lane & 15;
    const int ks = (lane >> 4) * 2;
    const int mbase = (lane >> 4) * 8;

    v8f acc[8];
    #pragma unroll
    for (int nt = 0; nt < 8; ++nt) acc[nt] = (v8f){};

    for (int k0 = 0; k0 < Cc; k0 += 4) {
        v2f af;                              // A[m=j][k=c] = Ts[k][m]
        af.x = Ts[k0 + ks][jb + nl];
        af.y = Ts[k0 + ks + 1][jb + nl];
        v2f bf[8];
        #pragma unroll
        for (int nt = 0; nt < 8; ++nt)
            bf[nt] = *(const v2f*)(outw + (size_t)(nt * 16 + nl) * Cc + k0 + ks);
        #pragma unroll
        for (int nt = 0; nt < 8; ++nt)
            acc[nt] = wmma4(af, bf[nt], acc[nt]);
    }

    // epilogue: * sigmoid-gate, store (i,j,c) with c contiguous across lanes
    #pragma unroll
    for (int nt = 0; nt < 8; ++nt) {
        int cout = nt * 16 + nl;
        #pragma unroll
        for (int r = 0; r < 8; ++r) {
            int j = j0 + jb + mbase + r;
            float val = acc[nt][r] * Gs[cout][jb + mbase + r];
            out[((size_t)i * Nn + j) * Cc + cout] = val;
        }
    }
}

// ---------------------------------------------------------------------------
extern "C" void kernel_launch(void* const* d_in, const int* in_sizes, int n_in,
                              void* d_out, int out_size, void* d_ws, size_t ws_size,
                              hipStream_t stream)
{
    const float* pair    = (const float*)d_in[0];
    const float* mask    = (const float*)d_in[1];
    const float* ln1w    = (const float*)d_in[2];
    const float* ln1b    = (const float*)d_in[3];
    const float* projw   = (const float*)d_in[4];
    const float* gatew   = (const float*)d_in[5];
    const float* ln2w    = (const float*)d_in[6];
    const float* ln2b    = (const float*)d_in[7];
    const float* outw    = (const float*)d_in[8];
    const float* gatingw = (const float*)d_in[9];

    float* a_ws = (float*)d_ws;
    float* b_ws = a_ws + (size_t)Cc * NP;
    float* g_ws = b_ws + (size_t)Cc * NP;
    float* tri  = g_ws + (size_t)Cc * NP;
    float* wcat = tri  + (size_t)Cc * NP;    // 640*128 floats = 320 KB

    k0_pack<<<640 * Cc / 4 / 256, 256, 0, stream>>>(projw, gatew, gatingw, wcat);
    k1_ln_proj<<<NP / 16, 128, 0, stream>>>(pair, mask, ln1w, ln1b, wcat,
                                            a_ws, b_ws, g_ws);
    k2_tri<<<dim3(Nn / 64, Nn / 64, Cc), 128, 0, stream>>>(a_ws, b_ws, tri);
    k3_out<<<dim3(Nn, Nn / 64), 128, 0, stream>>>(tri, g_ws, ln2w, ln2b, outw,
                                                  (float*)d_out);
}